// MultiHeadAttention_1580547970428
// MI455X (gfx1250) — compile-verified
//
#include <hip/hip_runtime.h>
#include <hip/hip_bf16.h>

#define B_  4
#define L_  1024
#define DM  512
#define NH  8
#define DK  64
#define DV  64

typedef __attribute__((ext_vector_type(2))) float v2f;
typedef __attribute__((ext_vector_type(8))) float v8f;

// D = A(16x4 f32) * B(4x16 f32) + C  -- full fp32 WMMA (matches fp32 reference)
__device__ __forceinline__ v8f wmma4(v2f a, v2f b, v8f c) {
  return __builtin_amdgcn_wmma_f32_16x16x4_f32(
      /*neg_a=*/false, a, /*neg_b=*/false, b,
      /*c_mod=*/(short)0, c, /*reuse_a=*/false, /*reuse_b=*/false);
}

// ---------------------------------------------------------------------------
// 1) Fused QKV projection: [4096,512] @ [512,512] -> head-split [B,H,L,64].
//    matrix 0 (Q) is pre-scaled by 1/sqrt(DK) = 0.125.
// ---------------------------------------------------------------------------
__global__ __launch_bounds__(256) void proj_kernel(
    const float* __restrict__ q, const float* __restrict__ k, const float* __restrict__ v,
    const float* __restrict__ wq, const float* __restrict__ wk, const float* __restrict__ wv,
    float* __restrict__ qs, float* __restrict__ kp, float* __restrict__ vp)
{
  const int wid  = threadIdx.x >> 5;
  const int lane = threadIdx.x & 31;
  const int half = lane >> 4;      // 0: K=k..k+1, 1: K=k+2..k+3
  const int lr   = lane & 15;

  const long gt  = (long)blockIdx.x * 8 + wid;       // 0 .. 3*256*32-1
  const int  mat = (int)(gt / (256 * 32));           // 0=Q 1=K 2=V
  const int  rem = (int)(gt % (256 * 32));
  const int  tm  = rem >> 5;                         // 0..255 (M tile)
  const int  tn  = rem & 31;                         // 0..31  (N tile)

  const float* A = (mat == 0) ? q  : (mat == 1) ? k  : v;
  const float* W = (mat == 0) ? wq : (mat == 1) ? wk : wv;
  float*       O = (mat == 0) ? qs : (mat == 1) ? kp : vp;
  const float scale = (mat == 0) ? 0.125f : 1.0f;

  const int m0 = tm * 16, n0 = tn * 16;
  const float* arow = A + (long)(m0 + lr) * DM;

  v8f acc = {};
  for (int k0 = 0; k0 < DM; k0 += 4) {
    const int kk = k0 + 2 * half;
    v2f a; a.x = arow[kk];                 a.y = arow[kk + 1];
    v2f b; b.x = W[(long)kk * DM + n0 + lr]; b.y = W[(long)(kk + 1) * DM + n0 + lr];
    acc = wmma4(a, b, acc);
  }

  const int c = n0 + lr, h = c >> 6, d = c & 63;
#pragma unroll
  for (int i = 0; i < 8; ++i) {
    const int m = m0 + i + 8 * half;       // 0..4095 = b*1024 + l
    const int b = m >> 10, l = m & 1023;
    O[(((long)(b * NH + h) * L_ + l) * DK) + d] = acc[i] * scale;
  }
}

// ---------------------------------------------------------------------------
// 2) Scores: S[b,h] = Qs[b,h] (1024x64) @ K[b,h]^T  -> attn region (raw)
// ---------------------------------------------------------------------------
__global__ __launch_bounds__(256) void qk_kernel(
    const float* __restrict__ qs, const float* __restrict__ kp, float* __restrict__ attn)
{
  const int wid = threadIdx.x >> 5, lane = threadIdx.x & 31;
  const int half = lane >> 4, lr = lane & 15;

  const long gt = (long)blockIdx.x * 8 + wid;        // 0..131071
  const int bh  = (int)(gt >> 12);
  const int rem = (int)(gt & 4095);
  const int qt  = rem >> 6, kt = rem & 63;

  const float* Q = qs + (long)bh * L_ * DK;
  const float* K = kp + (long)bh * L_ * DK;
  const int m0 = qt * 16, n0 = kt * 16;
  const float* qrow = Q + (long)(m0 + lr) * DK;
  const float* krow = K + (long)(n0 + lr) * DK;      // B[k][n] = K[n][k]

  v8f acc = {};
#pragma unroll
  for (int k0 = 0; k0 < DK; k0 += 4) {
    const int kk = k0 + 2 * half;
    v2f a; a.x = qrow[kk]; a.y = qrow[kk + 1];
    v2f b; b.x = krow[kk]; b.y = krow[kk + 1];
    acc = wmma4(a, b, acc);
  }

  float* S = attn + (long)bh * L_ * L_;
#pragma unroll
  for (int i = 0; i < 8; ++i) {
    const int m = m0 + i + 8 * half;
    S[(long)m * L_ + n0 + lr] = acc[i];
  }
}

// ---------------------------------------------------------------------------
// 3) Row softmax over 1024, one wave per row, register-resident.
// ---------------------------------------------------------------------------
__global__ __launch_bounds__(256) void softmax_kernel(float* __restrict__ attn)
{
  const int wid = threadIdx.x >> 5, lane = threadIdx.x & 31;
  const long row = (long)blockIdx.x * 8 + wid;       // 0..32767
  float* p = attn + row * L_;

  float4 vb[8];
  float mx = -3.402823e38f;
#pragma unroll
  for (int i = 0; i < 8; ++i) {
    vb[i] = ((const float4*)p)[lane + i * 32];
    mx = fmaxf(mx, fmaxf(fmaxf(vb[i].x, vb[i].y), fmaxf(vb[i].z, vb[i].w)));
  }
#pragma unroll
  for (int off = 16; off; off >>= 1) mx = fmaxf(mx, __shfl_xor(mx, off, 32));

  float sum = 0.f;
#pragma unroll
  for (int i = 0; i < 8; ++i) {
    vb[i].x = __expf(vb[i].x - mx); vb[i].y = __expf(vb[i].y - mx);
    vb[i].z = __expf(vb[i].z - mx); vb[i].w = __expf(vb[i].w - mx);
    sum += vb[i].x + vb[i].y + vb[i].z + vb[i].w;
  }
#pragma unroll
  for (int off = 16; off; off >>= 1) sum += __shfl_xor(sum, off, 32);

  const float inv = 1.0f / sum;
#pragma unroll
  for (int i = 0; i < 8; ++i) {
    vb[i].x *= inv; vb[i].y *= inv; vb[i].z *= inv; vb[i].w *= inv;
    ((float4*)p)[lane + i * 32] = vb[i];
  }
}

// ---------------------------------------------------------------------------
// 4) O[b,h] = P[b,h] (1024x1024) @ V[b,h] (1024x64)
// ---------------------------------------------------------------------------
__global__ __launch_bounds__(256) void pv_kernel(
    const float* __restrict__ attn, const float* __restrict__ vp, float* __restrict__ o)
{
  const int wid = threadIdx.x >> 5, lane = threadIdx.x & 31;
  const int half = lane >> 4, lr = lane & 15;

  const long gt = (long)blockIdx.x * 8 + wid;        // 0..8191
  const int bh  = (int)(gt >> 8);
  const int rem = (int)(gt & 255);
  const int qt  = rem >> 2, dt = rem & 3;

  const float* P = attn + (long)bh * L_ * L_;
  const float* V = vp + (long)bh * L_ * DV;
  const int m0 = qt * 16, n0 = dt * 16;
  const float* prow = P + (long)(m0 + lr) * L_;

  v8f acc = {};
#pragma unroll 4
  for (int k0 = 0; k0 < L_; k0 += 4) {
    const int kk = k0 + 2 * half;
    v2f a; a.x = prow[kk];                  a.y = prow[kk + 1];
    v2f b; b.x = V[(long)kk * DV + n0 + lr]; b.y = V[(long)(kk + 1) * DV + n0 + lr];
    acc = wmma4(a, b, acc);
  }

  float* orow = o + (long)bh * L_ * DV;
#pragma unroll
  for (int i = 0; i < 8; ++i)
    orow[(long)(m0 + i + 8 * half) * DV + n0 + lr] = acc[i];
}

// ---------------------------------------------------------------------------
// 5a) s[b,d] = mean over q of sum over h of o[b,h,q,d]
// ---------------------------------------------------------------------------
__global__ __launch_bounds__(256) void pool_kernel(
    const float* __restrict__ o, float* __restrict__ s)
{
  const int bd = blockIdx.x;                 // 0..255
  const int b = bd >> 6, d = bd & 63;
  const float* base = o + (long)b * NH * L_ * DV + d;
  float acc = 0.f;
  for (int i = threadIdx.x; i < NH * L_; i += 256) acc += base[(long)i * DV];
  __shared__ float sm[256];
  sm[threadIdx.x] = acc; __syncthreads();
  for (int st = 128; st; st >>= 1) {
    if (threadIdx.x < st) sm[threadIdx.x] += sm[threadIdx.x + st];
    __syncthreads();
  }
  if (threadIdx.x == 0) s[bd] = sm[0] * (1.0f / (float)L_);
}

// ---------------------------------------------------------------------------
// 5b) g[b,h,d] = softmax over h of (s[b,:] @ w_sk)[h,d]
// ---------------------------------------------------------------------------
__global__ __launch_bounds__(256) void gate_kernel(
    const float* __restrict__ s, const float* __restrict__ wsk, float* __restrict__ g)
{
  const int t = threadIdx.x;                 // one block of 256: (b,d)
  const int b = t >> 6, d = t & 63;
  float pre[NH];
#pragma unroll
  for (int h = 0; h < NH; ++h) {
    float acc = 0.f;
    for (int dv = 0; dv < DV; ++dv)
      acc += s[b * DV + dv] * wsk[dv * (NH * DV) + h * DV + d];
    pre[h] = acc;
  }
  float mx = pre[0];
#pragma unroll
  for (int h = 1; h < NH; ++h) mx = fmaxf(mx, pre[h]);
  float sum = 0.f;
#pragma unroll
  for (int h = 0; h < NH; ++h) { pre[h] = __expf(pre[h] - mx); sum += pre[h]; }
  const float inv = 1.0f / sum;
#pragma unroll
  for (int h = 0; h < NH; ++h) g[(b * NH + h) * DV + d] = pre[h] * inv;
}

// ---------------------------------------------------------------------------
// 5c) qo[b,q,d] = sum_h o[b,h,q,d] * g[b,h,d]
// ---------------------------------------------------------------------------
__global__ __launch_bounds__(256) void headsum_kernel(
    const float* __restrict__ o, const float* __restrict__ g, float* __restrict__ qo)
{
  const long idx = (long)blockIdx.x * 256 + threadIdx.x;   // B*L*DV
  const int d = (int)(idx & 63);
  const long bq = idx >> 6;
  const int b = (int)(bq >> 10), ql = (int)(bq & 1023);
  float acc = 0.f;
#pragma unroll
  for (int h = 0; h < NH; ++h)
    acc += o[(((long)(b * NH + h) * L_) + ql) * DV + d] * g[(b * NH + h) * DV + d];
  qo[idx] = acc;
}

// ---------------------------------------------------------------------------
// 6) out = LayerNorm(qo @ w_fc + residual); one block per (b,q) row.
// ---------------------------------------------------------------------------
__global__ __launch_bounds__(256) void final_kernel(
    const float* __restrict__ qo, const float* __restrict__ wfc,
    const float* __restrict__ resid, const float* __restrict__ gamma,
    const float* __restrict__ beta, float* __restrict__ out)
{
  const int row = blockIdx.x;                // 0..4095
  const int t = threadIdx.x;
  __shared__ float sq[DV];
  __shared__ float red[256];
  if (t < DV) sq[t] = qo[(long)row * DV + t];
  __syncthreads();

  float vals[2];
#pragma unroll
  for (int j = 0; j < 2; ++j) {
    const int col = t + j * 256;
    float acc = resid[(long)row * DM + col];
    for (int d = 0; d < DV; ++d) acc += sq[d] * wfc[d * DM + col];
    vals[j] = acc;
  }

  red[t] = vals[0] + vals[1]; __syncthreads();
  for (int st = 128; st; st >>= 1) { if (t < st) red[t] += red[t + st]; __syncthreads(); }
  const float mu = red[0] * (1.0f / (float)DM);
  __syncthreads();

  const float d0 = vals[0] - mu, d1 = vals[1] - mu;
  red[t] = d0 * d0 + d1 * d1; __syncthreads();
  for (int st = 128; st; st >>= 1) { if (t < st) red[t] += red[t + st]; __syncthreads(); }
  const float rstd = rsqrtf(red[0] * (1.0f / (float)DM) + 1e-6f);

#pragma unroll
  for (int j = 0; j < 2; ++j) {
    const int col = t + j * 256;
    out[(long)row * DM + col] = (vals[j] - mu) * rstd * gamma[col] + beta[col];
  }
}

// ---------------------------------------------------------------------------
extern "C" void kernel_launch(void* const* d_in, const int* in_sizes, int n_in,
                              void* d_out, int out_size, void* d_ws, size_t ws_size,
                              hipStream_t stream) {
  (void)in_sizes; (void)n_in; (void)out_size; (void)ws_size;
  const float* q     = (const float*)d_in[0];
  const float* k     = (const float*)d_in[1];
  const float* v     = (const float*)d_in[2];
  const float* w_qs  = (const float*)d_in[3];
  const float* w_ks  = (const float*)d_in[4];
  const float* w_vs  = (const float*)d_in[5];
  const float* w_sk  = (const float*)d_in[6];
  const float* w_fc  = (const float*)d_in[7];
  const float* ln_g  = (const float*)d_in[8];
  const float* ln_b  = (const float*)d_in[9];

  float* out  = (float*)d_out;                           // [4,1024,512]
  float* attn = out + (size_t)B_ * L_ * DM;              // [4,8,1024,1024]

  const size_t proj_elems = (size_t)B_ * NH * L_ * DK;   // 2,097,152
  float* qs = (float*)d_ws;
  float* kp = qs + proj_elems;
  float* vp = kp + proj_elems;
  float* o  = vp + proj_elems;
  float* s  = o  + proj_elems;                           // 256
  float* g  = s  + 256;                                  // 2048
  float* qo = g  + 2048;                                 // 262,144

  proj_kernel   <<<3072, 256, 0, stream>>>(q, k, v, w_qs, w_ks, w_vs, qs, kp, vp);
  qk_kernel     <<<16384, 256, 0, stream>>>(qs, kp, attn);
  softmax_kernel<<<4096, 256, 0, stream>>>(attn);
  pv_kernel     <<<1024, 256, 0, stream>>>(attn, vp, o);
  pool_kernel   <<<256, 256, 0, stream>>>(o, s);
  gate_kernel   <<<1, 256, 0, stream>>>(s, w_sk, g);
  headsum_kernel<<<1024, 256, 0, stream>>>(o, g, qo);
  final_kernel  <<<4096, 256, 0, stream>>>(qo, w_fc, q, ln_g, ln_b, out);
}